// ImageAdaptive3DModel_6416681140996
// MI455X (gfx1250) — compile-verified
//
#include <hip/hip_runtime.h>
#include <math.h>

typedef __attribute__((ext_vector_type(2))) float v2f;
typedef __attribute__((ext_vector_type(8))) float v8f;

#define LUT_DIM 33
#define LUT_DIM2 (33*33)
#define LUT_DIM3 (33*33*33)

__device__ __forceinline__ float lrelu_f(float v) { return v >= 0.f ? v : 0.2f * v; }

// ---------------------------------------------------------------------------
// Separable antialiased bilinear resize (triangle filter scaled by the
// downsample factor, matching jax.image.resize(antialias=True) semantics).
// ---------------------------------------------------------------------------
__global__ void resize_h_kernel(const float* __restrict__ in, float* __restrict__ out,
                                int H, int Win, int Wout, float scale)
{
    int i = blockIdx.x * blockDim.x + threadIdx.x;
    int total = 3 * H * Wout;
    if (i >= total) return;
    int ox  = i % Wout;
    int row = i / Wout;                       // c*H + y
    float sample = (ox + 0.5f) * scale - 0.5f;
    int j0 = (int)ceilf(sample - scale);
    int j1 = (int)floorf(sample + scale);
    const float* src = in + (size_t)row * Win;
    float acc = 0.f, wsum = 0.f;
    for (int j = j0; j <= j1; ++j) {
        float wt = 1.f - fabsf((float)j - sample) / scale;
        if (wt <= 0.f) continue;
        int jc = j < 0 ? 0 : (j >= Win ? Win - 1 : j);
        acc  += wt * src[jc];
        wsum += wt;
    }
    out[i] = acc / wsum;
}

__global__ void resize_v_kernel(const float* __restrict__ in, float* __restrict__ out,
                                int Hin, int Hout, int W, float scale)
{
    int i = blockIdx.x * blockDim.x + threadIdx.x;
    int total = 3 * Hout * W;
    if (i >= total) return;
    int ox = i % W;
    int t  = i / W;
    int oy = t % Hout;
    int c  = t / Hout;
    float sample = (oy + 0.5f) * scale - 0.5f;
    int j0 = (int)ceilf(sample - scale);
    int j1 = (int)floorf(sample + scale);
    float acc = 0.f, wsum = 0.f;
    for (int j = j0; j <= j1; ++j) {
        float wt = 1.f - fabsf((float)j - sample) / scale;
        if (wt <= 0.f) continue;
        int jc = j < 0 ? 0 : (j >= Hin ? Hin - 1 : j);
        acc  += wt * in[((size_t)c * Hin + jc) * W + ox];
        wsum += wt;
    }
    out[i] = acc / wsum;
}

// ---------------------------------------------------------------------------
// 3x3 stride-2 pad-1 conv as implicit GEMM with V_WMMA_F32_16X16X4_F32.
// One wave computes a 16(cout) x 16(pixel) tile.  Each input channel is given
// a padded K-stride of 16 (9 real taps + 7 zeros), so the 8 K-slots this lane
// feeds (t = 4j + 2*half + v) are FIXED across channels.  All tap decomposition
// (ky/kx, clamped offsets, validity masks) is hoisted into a preamble; the
// inner per-channel loop is 16 unconditional loads + 4 WMMAs — no divisions,
// no divergent branches, so loads clause and overlap with the WMMA chain.
// Lane layouts per ISA: A(16x4): m=lane&15, a[v] = K(2*half+v of chunk);
// B(4x16): n=lane&15, same K slot; D: VGPR r -> M = r + 8*half, N = lane&15.
// EXEC is full at the WMMA (32-thread block, uniform control flow).
// ---------------------------------------------------------------------------
__global__ void conv3x3_s2_wmma(const float* __restrict__ x, const float* __restrict__ w,
                                const float* __restrict__ bias, float* __restrict__ y,
                                int Cin, int Cout, int Hin, int Win, int Hout, int Wout,
                                int fuse_lrelu)
{
    const int lane = threadIdx.x;       // 0..31 (wave32)
    const int half = lane >> 4;
    const int l    = lane & 15;
    const int pixTiles = (Hout * Wout) >> 4;
    const int ct = blockIdx.x / pixTiles;
    const int pt = blockIdx.x - ct * pixTiles;

    const int K    = Cin * 9;           // real K (weight layout stride)
    const int HW   = Hin * Win;
    const int pix  = (pt << 4) + l;
    const int oy   = pix / Wout;
    const int ox   = pix - oy * Wout;
    const int co_a = (ct << 4) + l;     // A-matrix row (cout) held by this lane

    // ---- preamble: fixed taps for this lane (4 chunks x 2 values) ----
    int   woff[4][2];  float wmsk[4][2];
    int   boff[4][2];  float bmsk[4][2];
#pragma unroll
    for (int j = 0; j < 4; ++j) {
#pragma unroll
        for (int v = 0; v < 2; ++v) {
            int  t   = 4 * j + 2 * half + v;   // 0..15 within padded K-chunk
            bool tv  = (t < 9);
            int  tc  = tv ? t : 0;
            int  ky  = tc / 3;
            int  kx  = tc - ky * 3;
            int  iy  = 2 * oy - 1 + ky;
            int  ix  = 2 * ox - 1 + kx;
            bool inb = tv && (iy >= 0) && (iy < Hin) && (ix >= 0) && (ix < Win);
            int  iyc = iy < 0 ? 0 : (iy >= Hin ? Hin - 1 : iy);
            int  ixc = ix < 0 ? 0 : (ix >= Win ? Win - 1 : ix);
            woff[j][v] = tc;                 // + ci*9 per channel
            wmsk[j][v] = tv  ? 1.f : 0.f;
            boff[j][v] = iyc * Win + ixc;    // + ci*HW per channel
            bmsk[j][v] = inb ? 1.f : 0.f;
        }
    }

    v8f acc = {0.f, 0.f, 0.f, 0.f, 0.f, 0.f, 0.f, 0.f};

    const float* wr = w + (size_t)co_a * K;  // this lane's weight row
    for (int ci = 0; ci < Cin; ++ci) {
        const float* wp = wr + ci * 9;
        const float* xp = x + (size_t)ci * HW;
#pragma unroll
        for (int j = 0; j < 4; ++j) {
            v2f afrag, bfrag;
#pragma unroll
            for (int v = 0; v < 2; ++v) {
                afrag[v] = wmsk[j][v] * wp[woff[j][v]];
                bfrag[v] = bmsk[j][v] * xp[boff[j][v]];
            }
            acc = __builtin_amdgcn_wmma_f32_16x16x4_f32(
                /*neg_a=*/false, afrag, /*neg_b=*/false, bfrag,
                /*c_mod=*/(short)0, acc, /*reuse_a=*/false, /*reuse_b=*/false);
        }
    }

#pragma unroll
    for (int r = 0; r < 8; ++r) {
        int co = (ct << 4) + r + 8 * half;
        float val = acc[r] + bias[co];
        if (fuse_lrelu) val = lrelu_f(val);
        y[((size_t)co * Hout + oy) * Wout + ox] = val;
    }
}

// ---------------------------------------------------------------------------
// Fused lrelu + instance norm, in place. One block per channel.
// ---------------------------------------------------------------------------
__global__ void lrelu_inorm_kernel(float* __restrict__ x, const float* __restrict__ g,
                                   const float* __restrict__ be, int HW)
{
    __shared__ float ss[256];
    __shared__ float ss2[256];
    const int c = blockIdx.x;
    float* p = x + (size_t)c * HW;

    float s = 0.f, s2 = 0.f;
    for (int i = threadIdx.x; i < HW; i += blockDim.x) {
        float v = lrelu_f(p[i]);
        s  += v;
        s2 += v * v;
    }
    ss[threadIdx.x]  = s;
    ss2[threadIdx.x] = s2;
    __syncthreads();
    for (int o = blockDim.x >> 1; o > 0; o >>= 1) {
        if (threadIdx.x < (unsigned)o) {
            ss[threadIdx.x]  += ss[threadIdx.x + o];
            ss2[threadIdx.x] += ss2[threadIdx.x + o];
        }
        __syncthreads();
    }
    float mean = ss[0] / (float)HW;
    float var  = ss2[0] / (float)HW - mean * mean;
    float inv  = rsqrtf(var + 1e-5f);
    float gg   = g[c] * inv;
    float bb   = be[c];
    for (int i = threadIdx.x; i < HW; i += blockDim.x) {
        float v = lrelu_f(p[i]);
        p[i] = (v - mean) * gg + bb;
    }
}

// ---------------------------------------------------------------------------
// Head conv: (3,128,8,8) over (128,8,8), pad 0 -> flat dot of 8192 per output.
// ---------------------------------------------------------------------------
__global__ void conv6_pred_kernel(const float* __restrict__ x5, const float* __restrict__ w6,
                                  const float* __restrict__ b6, float* __restrict__ pred)
{
    __shared__ float ss[256];
    const int c = blockIdx.x;  // 0..2
    float s = 0.f;
    for (int i = threadIdx.x; i < 128 * 64; i += blockDim.x)
        s += x5[i] * w6[(size_t)c * 8192 + i];
    ss[threadIdx.x] = s;
    __syncthreads();
    for (int o = blockDim.x >> 1; o > 0; o >>= 1) {
        if (threadIdx.x < (unsigned)o) ss[threadIdx.x] += ss[threadIdx.x + o];
        __syncthreads();
    }
    if (threadIdx.x == 0) pred[c] = ss[0] + b6[c];
}

// ---------------------------------------------------------------------------
// final_lut = pred[0]*lut0 + pred[1]*lut1 + pred[2]*lut2
// ---------------------------------------------------------------------------
__global__ void combine_lut_kernel(const float* __restrict__ l0, const float* __restrict__ l1,
                                   const float* __restrict__ l2, const float* __restrict__ pred,
                                   float* __restrict__ out)
{
    int i = blockIdx.x * blockDim.x + threadIdx.x;
    if (i >= 3 * LUT_DIM3) return;
    out[i] = pred[0] * l0[i] + pred[1] * l1[i] + pred[2] * l2[i];
}

// ---------------------------------------------------------------------------
// Trilinear LUT apply: 4 pixels per thread, float4 plane loads/stores.
// 431 KB LUT stays hot in L2; gathers are latency-hidden by occupancy.
// ---------------------------------------------------------------------------
__device__ __forceinline__ void lut_apply_px(float r, float g, float b,
                                             const float* __restrict__ lut, float* o)
{
    float xr = r * 32.f, xg = g * 32.f, xb = b * 32.f;
    int ir = (int)floorf(xr); ir = ir < 0 ? 0 : (ir > 31 ? 31 : ir);
    int ig = (int)floorf(xg); ig = ig < 0 ? 0 : (ig > 31 ? 31 : ig);
    int ib = (int)floorf(xb); ib = ib < 0 ? 0 : (ib > 31 ? 31 : ib);
    float fr = xr - (float)ir;
    float fg = xg - (float)ig;
    float fb = xb - (float)ib;
    int base = (ib * LUT_DIM + ig) * LUT_DIM + ir;
    float o0 = 0.f, o1 = 0.f, o2 = 0.f;
#pragma unroll
    for (int db = 0; db < 2; ++db) {
        float wb = db ? fb : 1.f - fb;
#pragma unroll
        for (int dg = 0; dg < 2; ++dg) {
            float wg = dg ? fg : 1.f - fg;
#pragma unroll
            for (int dr = 0; dr < 2; ++dr) {
                float wt = wb * wg * (dr ? fr : 1.f - fr);
                int idx = base + db * LUT_DIM2 + dg * LUT_DIM + dr;
                o0 += wt * lut[idx];
                o1 += wt * lut[LUT_DIM3 + idx];
                o2 += wt * lut[2 * LUT_DIM3 + idx];
            }
        }
    }
    o[0] = o0; o[1] = o1; o[2] = o2;
}

__global__ void trilerp_kernel(const float* __restrict__ img, const float* __restrict__ lut,
                               float* __restrict__ out, int P)
{
    int q  = blockIdx.x * blockDim.x + threadIdx.x;
    int i0 = q * 4;
    if (i0 >= P) return;
    float4 R = *(const float4*)(img + i0);
    float4 G = *(const float4*)(img + P + i0);
    float4 B = *(const float4*)(img + 2 * P + i0);
    float4 OR, OG, OB;
    float o[3];
    lut_apply_px(R.x, G.x, B.x, lut, o); OR.x = o[0]; OG.x = o[1]; OB.x = o[2];
    lut_apply_px(R.y, G.y, B.y, lut, o); OR.y = o[0]; OG.y = o[1]; OB.y = o[2];
    lut_apply_px(R.z, G.z, B.z, lut, o); OR.z = o[0]; OG.z = o[1]; OB.z = o[2];
    lut_apply_px(R.w, G.w, B.w, lut, o); OR.w = o[0]; OG.w = o[1]; OB.w = o[2];
    *(float4*)(out + i0)         = OR;
    *(float4*)(out + P + i0)     = OG;
    *(float4*)(out + 2 * P + i0) = OB;
}

// ---------------------------------------------------------------------------
// Launch
// ---------------------------------------------------------------------------
extern "C" void kernel_launch(void* const* d_in, const int* in_sizes, int n_in,
                              void* d_out, int out_size, void* d_ws, size_t ws_size,
                              hipStream_t stream)
{
    (void)in_sizes; (void)n_in; (void)out_size; (void)ws_size;

    const float* img  = (const float*)d_in[0];
    const float* lut0 = (const float*)d_in[1];
    const float* lut1 = (const float*)d_in[2];
    const float* lut2 = (const float*)d_in[3];
    const float* w1 = (const float*)d_in[4];
    const float* b1 = (const float*)d_in[5];
    const float* g1 = (const float*)d_in[6];
    const float* be1 = (const float*)d_in[7];
    const float* w2 = (const float*)d_in[8];
    const float* b2 = (const float*)d_in[9];
    const float* g2 = (const float*)d_in[10];
    const float* be2 = (const float*)d_in[11];
    const float* w3 = (const float*)d_in[12];
    const float* b3 = (const float*)d_in[13];
    const float* g3 = (const float*)d_in[14];
    const float* be3 = (const float*)d_in[15];
    const float* w4 = (const float*)d_in[16];
    const float* b4 = (const float*)d_in[17];
    const float* g4 = (const float*)d_in[18];
    const float* be4 = (const float*)d_in[19];
    const float* w5 = (const float*)d_in[20];
    const float* b5 = (const float*)d_in[21];
    const float* w6 = (const float*)d_in[22];
    const float* b6 = (const float*)d_in[23];

    const int H = 2160, W = 3840;

    // Workspace carve-up (floats)
    float* ws   = (float*)d_ws;
    size_t off  = 0;
    float* tmp1 = ws + off; off += (size_t)3 * H * 256;   // H-pass result [3][2160][256]
    float* x0   = ws + off; off += (size_t)3 * 256 * 256; // resized   [3][256][256]
    float* x1   = ws + off; off += (size_t)16 * 128 * 128;
    float* x2   = ws + off; off += (size_t)32 * 64 * 64;
    float* x3   = ws + off; off += (size_t)64 * 32 * 32;
    float* x4   = ws + off; off += (size_t)128 * 16 * 16;
    float* x5   = ws + off; off += (size_t)128 * 8 * 8;
    float* pred = ws + off; off += 4;
    float* flut = ws + off; off += (size_t)3 * LUT_DIM3;

    // 1) antialiased bilinear 4K -> 256x256 (separable)
    {
        int n = 3 * H * 256;
        resize_h_kernel<<<(n + 255) / 256, 256, 0, stream>>>(img, tmp1, H, W, 256, (float)W / 256.f);
    }
    {
        int n = 3 * 256 * 256;
        resize_v_kernel<<<(n + 255) / 256, 256, 0, stream>>>(tmp1, x0, H, 256, 256, (float)H / 256.f);
    }

    // 2) conv stack (WMMA implicit GEMM) + fused lrelu/instance-norm
    conv3x3_s2_wmma<<<1 * (128 * 128 / 16), 32, 0, stream>>>(x0, w1, b1, x1, 3, 16, 256, 256, 128, 128, 0);
    lrelu_inorm_kernel<<<16, 256, 0, stream>>>(x1, g1, be1, 128 * 128);

    conv3x3_s2_wmma<<<2 * (64 * 64 / 16), 32, 0, stream>>>(x1, w2, b2, x2, 16, 32, 128, 128, 64, 64, 0);
    lrelu_inorm_kernel<<<32, 256, 0, stream>>>(x2, g2, be2, 64 * 64);

    conv3x3_s2_wmma<<<4 * (32 * 32 / 16), 32, 0, stream>>>(x2, w3, b3, x3, 32, 64, 64, 64, 32, 32, 0);
    lrelu_inorm_kernel<<<64, 256, 0, stream>>>(x3, g3, be3, 32 * 32);

    conv3x3_s2_wmma<<<8 * (16 * 16 / 16), 32, 0, stream>>>(x3, w4, b4, x4, 64, 128, 32, 32, 16, 16, 0);
    lrelu_inorm_kernel<<<128, 256, 0, stream>>>(x4, g4, be4, 16 * 16);

    conv3x3_s2_wmma<<<8 * (8 * 8 / 16), 32, 0, stream>>>(x4, w5, b5, x5, 128, 128, 16, 16, 8, 8, 1);

    // 3) head -> 3 scalars
    conv6_pred_kernel<<<3, 256, 0, stream>>>(x5, w6, b6, pred);

    // 4) blend LUTs
    {
        int n = 3 * LUT_DIM3;
        combine_lut_kernel<<<(n + 255) / 256, 256, 0, stream>>>(lut0, lut1, lut2, pred, flut);
    }

    // 5) trilinear apply (bandwidth-bound main pass)
    {
        int P = H * W;  // 8,294,400 (multiple of 4)
        trilerp_kernel<<<(P / 4 + 255) / 256, 256, 0, stream>>>(img, flut, (float*)d_out, P);
    }
}